// OneLayerGCNEncoder_47820165874092
// MI455X (gfx1250) — compile-verified
//
#include <hip/hip_runtime.h>
#include <hip/hip_bf16.h>

typedef float v2f __attribute__((ext_vector_type(2)));
typedef float v8f __attribute__((ext_vector_type(8)));
typedef unsigned int v4u __attribute__((ext_vector_type(4)));
typedef int v4i __attribute__((ext_vector_type(4)));
typedef int v8i __attribute__((ext_vector_type(8)));

#define IN_DIM  512
#define OUT_DIM 128
#define LDS_STRIDE 144   // 128 dwords + 16 pad dwords (TDM pad) -> bank-conflict-free

// ---------------------------------------------------------------------------
// 1) deg[i] = 1.0 (self loop)
__global__ __launch_bounds__(256) void k_init_deg(float* __restrict__ deg, int n) {
    int i = blockIdx.x * blockDim.x + threadIdx.x;
    if (i < n) deg[i] = 1.0f;
}

// 2) deg[dst[e]] += 1
__global__ __launch_bounds__(256) void k_accum_deg(const long long* __restrict__ ei,
                                                   float* __restrict__ deg, int e_cnt) {
    int e = blockIdx.x * blockDim.x + threadIdx.x;
    if (e < e_cnt) {
        long long d = ei[(size_t)e_cnt + e];
        unsafeAtomicAdd(&deg[d], 1.0f);        // global_atomic_add_f32, L2-resident
    }
}

// 3) dinv[i] = deg>0 ? rsqrt(deg) : 0
__global__ __launch_bounds__(256) void k_dinv(const float* __restrict__ deg,
                                              float* __restrict__ dinv, int n) {
    int i = blockIdx.x * blockDim.x + threadIdx.x;
    if (i < n) {
        float dv = deg[i];
        dinv[i] = (dv > 0.0f) ? rsqrtf(dv) : 0.0f;
    }
}

// ---------------------------------------------------------------------------
// 4) H = X @ W. W (512x128 f32, 256KB) is DMA'd into LDS once per block by the
//    Tensor Data Mover with row padding (stride 144 dwords). Each wave computes
//    a 16x128 output strip: per K-step, one b64 A load from x (streamed once)
//    feeds 8 WMMAs whose B fragments come from LDS via the DS pipe.
__global__ __launch_bounds__(320) void k_gemm_wmma(const float* __restrict__ X,
                                                   const float* __restrict__ W,
                                                   float* __restrict__ H,
                                                   int nstrips) {
    __shared__ float Wlds[IN_DIM * LDS_STRIDE];   // 512*144*4 = 288 KB

    const int wave = threadIdx.x >> 5;
    const int lane = threadIdx.x & 31;

    if (wave == 0) {
        // ---- Tensor DMA descriptor: 2D tile, 128 x 512 elements of 4B ----
        unsigned long long ga = (unsigned long long)(uintptr_t)W;
        unsigned int lds_addr = (unsigned int)(uintptr_t)&Wlds[0];  // low 32 = LDS offset
        v4u g0;
        g0[0] = 1u;                                   // count = 1 valid descriptor
        g0[1] = lds_addr;                             // lds_addr
        g0[2] = (unsigned int)(ga & 0xffffffffu);     // global_addr[31:0]
        g0[3] = (unsigned int)((ga >> 32) & 0x1ffffffu) | 0x80000000u; // addr[56:32] | type=2
        v8i g1;
        // data_size=2(4B) | pad_enable | pad_interval=6(128dw) | pad_amount=15(16dw)
        g1[0] = (int)((2u << 16) | (1u << 20) | (6u << 22) | (15u << 25));
        g1[1] = (int)(128u << 16);                    // tensor_dim0 = 128 (row length)
        g1[2] = (int)(512u << 16);                    // tensor_dim1 = 512 (rows)
        g1[3] = (int)(128u << 16);                    // tile_dim0 = 128
        g1[4] = (int)512;                             // tile_dim1 = 512
        g1[5] = (int)128;                             // tensor_dim0_stride = 128
        g1[6] = 0;
        g1[7] = 0;
        v4i z4 = {0, 0, 0, 0};                        // groups 2/3: zeros (2D tensor)
        v8i z8 = {0, 0, 0, 0, 0, 0, 0, 0};
        __builtin_amdgcn_tensor_load_to_lds(g0, g1, z4, z4, z8, 0);
        __builtin_amdgcn_s_wait_tensorcnt(0);
    }
    __syncthreads();

    const int strip = blockIdx.x * 10 + wave;         // 16-row output strip
    if (strip < nstrips) {
        const int m  = lane & 15;                     // row (A) / col (B,D) within tile
        const int hi = lane >> 4;                     // 0: K pair {0,1}; 1: K pair {2,3}

        const float* __restrict__ xrow = X + (size_t)(strip * 16 + m) * IN_DIM;

        v8f acc[8] = {};
        for (int k0 = 0; k0 < IN_DIM; k0 += 4) {
            const int kk = k0 + hi * 2;
            v2f a = *(const v2f*)(xrow + kk);         // global b64, x streamed once
#pragma unroll
            for (int t = 0; t < 8; ++t) {
                const int idx = kk * LDS_STRIDE + t * 16 + m;
                v2f b;
                b.x = Wlds[idx];                      // DS pipe; conflict-free banks
                b.y = Wlds[idx + LDS_STRIDE];
                acc[t] = __builtin_amdgcn_wmma_f32_16x16x4_f32(
                    false, a, false, b, (short)0, acc[t], false, false);
            }
        }

        // D layout: VGPR r -> (M = r + hi*8, N = lane&15)
        const int mbase = strip * 16 + (hi << 3);
#pragma unroll
        for (int t = 0; t < 8; ++t)
#pragma unroll
            for (int r = 0; r < 8; ++r)
                H[(size_t)(mbase + r) * OUT_DIM + t * 16 + m] = acc[t][r];
    }
}

// ---------------------------------------------------------------------------
// 5) out[i,:] = H[i,:] * dinv[i]^2 + b   (self-loop term; initializes out)
__global__ __launch_bounds__(256) void k_self_loop(const float* __restrict__ H,
                                                   const float* __restrict__ dinv,
                                                   const float* __restrict__ bias,
                                                   float* __restrict__ out, size_t total) {
    size_t idx = (size_t)blockIdx.x * blockDim.x + threadIdx.x;
    if (idx < total) {
        size_t row = idx >> 7;
        int    c   = (int)(idx & 127);
        float  di  = dinv[row];
        out[idx] = H[idx] * (di * di) + bias[c];
    }
}

// 6) scatter: out[dst,:] += H[src,:] * (dinv[src]*dinv[dst])
//    One wave per edge; lane handles a float4 of the 128-wide row. H (51.2 MB)
//    is L2-resident (192 MB), so gathers and float atomics resolve in L2.
__global__ __launch_bounds__(256) void k_scatter(const long long* __restrict__ ei,
                                                 const float* __restrict__ H,
                                                 const float* __restrict__ dinv,
                                                 float* __restrict__ out, int e_cnt) {
    int warp = (int)((blockIdx.x * (size_t)blockDim.x + threadIdx.x) >> 5);
    int lane = threadIdx.x & 31;
    if (warp >= e_cnt) return;
    long long s = ei[warp];
    long long d = ei[(size_t)e_cnt + warp];
    float nrm = dinv[s] * dinv[d];
    const float4 hv = *(const float4*)(H + (size_t)s * OUT_DIM + lane * 4);
    float* o = out + (size_t)d * OUT_DIM + lane * 4;
    unsafeAtomicAdd(o + 0, hv.x * nrm);
    unsafeAtomicAdd(o + 1, hv.y * nrm);
    unsafeAtomicAdd(o + 2, hv.z * nrm);
    unsafeAtomicAdd(o + 3, hv.w * nrm);
}

// ---------------------------------------------------------------------------
extern "C" void kernel_launch(void* const* d_in, const int* in_sizes, int n_in,
                              void* d_out, int out_size, void* d_ws, size_t ws_size,
                              hipStream_t stream) {
    const float*     x  = (const float*)d_in[0];
    const long long* ei = (const long long*)d_in[1];   // int64 (2,E)
    const float*     W  = (const float*)d_in[2];
    const float*     b  = (const float*)d_in[3];
    float*           out = (float*)d_out;

    const int N = in_sizes[0] / IN_DIM;   // 100000
    const int E = in_sizes[1] / 2;        // 1600000

    float* ws   = (float*)d_ws;
    float* deg  = ws;                     // N floats
    float* dinv = ws + N;                 // N floats
    float* H    = ws + 2 * (size_t)N;     // N*128 floats

    // degrees + normalization
    k_init_deg <<<(N + 255) / 256, 256, 0, stream>>>(deg, N);
    k_accum_deg<<<(E + 255) / 256, 256, 0, stream>>>(ei, deg, E);
    k_dinv     <<<(N + 255) / 256, 256, 0, stream>>>(deg, dinv, N);

    // dense transform: H = X @ W (WMMA f32, TDM-staged W in LDS)
    const int nstrips = N / 16;                        // 6250
    const int blocks  = (nstrips + 9) / 10;            // 625 (exact)
    k_gemm_wmma<<<blocks, 320, 0, stream>>>(x, W, H, nstrips);

    // out = self-loop term + bias
    size_t total = (size_t)N * OUT_DIM;
    k_self_loop<<<(unsigned)((total + 255) / 256), 256, 0, stream>>>(H, dinv, b, out, total);

    // edge aggregation (wave per edge, L2-resident atomics)
    size_t threads = (size_t)E * 32;
    k_scatter<<<(unsigned)((threads + 255) / 256), 256, 0, stream>>>(ei, H, dinv, out, E);
}